// Attention_47459388621522
// MI455X (gfx1250) — compile-verified
//
#include <hip/hip_runtime.h>
#include <stdint.h>

// Problem constants
#define Bdim 256
#define Ndim 2048
#define Ddim 256
#define TwoD 512
#define NTILES (Bdim * (Ndim / 16))   // 32768 16-row tiles

typedef float v2f __attribute__((ext_vector_type(2)));
typedef float v4f __attribute__((ext_vector_type(4)));
typedef float v8f __attribute__((ext_vector_type(8)));

#define WSTRIDE 260   // 256 + pad: 260%64=4 -> conflict-free row access, 16B aligned rows
#define ASTRIDE 260

// ---------------------------------------------------------------------------
// Kernel 1: c[b][d] = sum_e W[d][256+e] * hid[b][e]   (tiny, L2-resident)
// ---------------------------------------------------------------------------
__global__ void precompute_c_kernel(const float* __restrict__ W,
                                    const float* __restrict__ hid,
                                    float* __restrict__ c) {
    const int b = blockIdx.x;
    const int d = threadIdx.x;
    __shared__ float lh[Ddim];
    lh[d] = hid[(size_t)b * Ddim + d];
    __syncthreads();
    const float* wrow = W + (size_t)d * TwoD + Ddim;  // second half of row d
    float acc = 0.f;
    #pragma unroll 4
    for (int e = 0; e < Ddim; e += 4) {
        v4f wv = *(const v4f*)(wrow + e);
        acc += wv.x * lh[e] + wv.y * lh[e + 1] + wv.z * lh[e + 2] + wv.w * lh[e + 3];
    }
    c[(size_t)b * Ddim + d] = acc;
}

// ---------------------------------------------------------------------------
// Kernel 2: scores[b][n] = sum_d v[d]*tanh( sum_e W[d][e]*feature[b,n,e] + c[b][d] )
// WMMA f32 16x16x4; W1 resident in 260KiB LDS; feature tiles double-buffered
// via GLOBAL_LOAD_ASYNC_TO_LDS_B128 (ASYNCcnt) to overlap HBM with WMMA.
// ---------------------------------------------------------------------------
__global__ void attn_scores_kernel(const float* __restrict__ feature,
                                   const float* __restrict__ W,
                                   const float* __restrict__ vvec,
                                   const float* __restrict__ cbd,
                                   float* __restrict__ scores) {
    __shared__ float lW[Ddim * WSTRIDE];     // 266,240 B : W1 [d][e]
    __shared__ float lA[2][16 * ASTRIDE];    //  33,280 B : double-buffered tile [m][e]
    __shared__ float lSc[8][16];             //     512 B : per-wave partial scores

    const int tid  = threadIdx.x;
    const int lane = tid & 31;
    const int wvid = tid >> 5;          // wave id 0..7

    // --- Load W1 (first 256 cols of each 512-wide row) into LDS, once ---
    for (int i = tid; i < Ddim * 64; i += 256) {
        const int d = i >> 6;
        const int q = i & 63;
        v4f val = *(const v4f*)(W + (size_t)d * TwoD + q * 4);
        *(v4f*)(&lW[d * WSTRIDE + q * 4]) = val;
    }
    // (visibility covered by the barrier inside the tile loop)

    // per-thread async-copy coordinates: 64 contiguous bytes of one tile row
    const int ar  = tid >> 4;            // row 0..15
    const int ac0 = (tid & 15) << 4;     // col 0..240 step 16 floats

    const int col   = lane & 15;         // d-column within 16-wide tile / A row id
    const int ehalf = (lane >> 4) << 1;  // lanes 0-15: K=+0,+1 ; lanes 16-31: K=+2,+3

    // prologue: kick off async load of the first tile into buffer 0
    int cur = 0;
    {
        const int t0 = blockIdx.x;
        const int b  = t0 >> 7;
        const int n0 = (t0 & 127) << 4;
        uint32_t lds = (uint32_t)(uintptr_t)&lA[0][ar * ASTRIDE + ac0];
        uint64_t ga  = (uint64_t)(uintptr_t)(feature +
                        ((size_t)(b * Ndim + n0 + ar) << 8) + ac0);
        asm volatile(
            "global_load_async_to_lds_b128 %0, %1, off\n\t"
            "global_load_async_to_lds_b128 %0, %1, off offset:16\n\t"
            "global_load_async_to_lds_b128 %0, %1, off offset:32\n\t"
            "global_load_async_to_lds_b128 %0, %1, off offset:48"
            :: "v"(lds), "v"(ga) : "memory");
    }

    for (int t = blockIdx.x; t < NTILES; t += gridDim.x) {
        const int b  = t >> 7;
        const int n0 = (t & 127) << 4;

        // wait for this block's async tile, then make it visible to all waves
        asm volatile("s_wait_asynccnt 0x0" ::: "memory");
        __syncthreads();

        // prefetch next tile into the other buffer while we compute
        const int tn = t + gridDim.x;
        if (tn < NTILES) {
            const int bn  = tn >> 7;
            const int nn0 = (tn & 127) << 4;
            uint32_t lds = (uint32_t)(uintptr_t)&lA[cur ^ 1][ar * ASTRIDE + ac0];
            uint64_t ga  = (uint64_t)(uintptr_t)(feature +
                            ((size_t)(bn * Ndim + nn0 + ar) << 8) + ac0);
            asm volatile(
                "global_load_async_to_lds_b128 %0, %1, off\n\t"
                "global_load_async_to_lds_b128 %0, %1, off offset:16\n\t"
                "global_load_async_to_lds_b128 %0, %1, off offset:32\n\t"
                "global_load_async_to_lds_b128 %0, %1, off offset:48"
                :: "v"(lds), "v"(ga) : "memory");
        }

        // --- each wave: two 16-wide d-tiles ---
        const int d0 = wvid * 32;
        const int d1 = d0 + 16;
        v8f acc0 = {0.f,0.f,0.f,0.f,0.f,0.f,0.f,0.f};
        v8f acc1 = {0.f,0.f,0.f,0.f,0.f,0.f,0.f,0.f};

        const float* aBase  = &lA[cur][col * ASTRIDE + ehalf];
        const float* b0Base = &lW[(d0 + col) * WSTRIDE + ehalf];
        const float* b1Base = &lW[(d1 + col) * WSTRIDE + ehalf];

        #pragma unroll 8
        for (int k = 0; k < 64; ++k) {
            v2f a   = *(const v2f*)(aBase  + 4 * k);
            v2f bb0 = *(const v2f*)(b0Base + 4 * k);
            v2f bb1 = *(const v2f*)(b1Base + 4 * k);
            acc0 = __builtin_amdgcn_wmma_f32_16x16x4_f32(false, a, false, bb0,
                                                         (short)0, acc0, false, false);
            acc1 = __builtin_amdgcn_wmma_f32_16x16x4_f32(false, a, false, bb1,
                                                         (short)0, acc1, false, false);
        }

        // --- epilogue: tanh, scale by v, reduce over d ---
        const float vd0 = vvec[d0 + col];
        const float vd1 = vvec[d1 + col];
        const float cv0 = cbd[(size_t)b * Ddim + d0 + col];
        const float cv1 = cbd[(size_t)b * Ddim + d1 + col];
        float sc[8];
        #pragma unroll
        for (int r = 0; r < 8; ++r)
            sc[r] = vd0 * tanhf(acc0[r] + cv0) + vd1 * tanhf(acc1[r] + cv1);

        // reduce across the 16 lanes of each half-wave (columns of this wave)
        #pragma unroll
        for (int m = 1; m < 16; m <<= 1)
            #pragma unroll
            for (int r = 0; r < 8; ++r)
                sc[r] += __shfl_xor(sc[r], m, 32);

        if (lane == 0) {
            #pragma unroll
            for (int r = 0; r < 8; ++r) lSc[wvid][r] = sc[r];       // rows 0..7
        } else if (lane == 16) {
            #pragma unroll
            for (int r = 0; r < 8; ++r) lSc[wvid][8 + r] = sc[r];   // rows 8..15
        }
        __syncthreads();

        // deterministic cross-wave sum (fixed order), write 16 scores
        if (tid < 16) {
            float s = 0.f;
            #pragma unroll
            for (int w2 = 0; w2 < 8; ++w2) s += lSc[w2][tid];
            scores[(size_t)b * Ndim + n0 + tid] = s;
        }
        cur ^= 1;
    }
}

// ---------------------------------------------------------------------------
// Kernel 3: softmax over n (2048) per batch row
// ---------------------------------------------------------------------------
__global__ void softmax_kernel(const float* __restrict__ scores,
                               float* __restrict__ out) {
    const int b = blockIdx.x;
    const int t = threadIdx.x;
    __shared__ float red[256];
    const float* row = scores + (size_t)b * Ndim;

    float vals[8];
    float m = -3.402823466e+38f;
    #pragma unroll
    for (int j = 0; j < 8; ++j) {
        vals[j] = row[t + 256 * j];
        m = fmaxf(m, vals[j]);
    }
    red[t] = m;
    __syncthreads();
    for (int s = 128; s > 0; s >>= 1) {
        if (t < s) red[t] = fmaxf(red[t], red[t + s]);
        __syncthreads();
    }
    m = red[0];
    __syncthreads();

    float sum = 0.f;
    #pragma unroll
    for (int j = 0; j < 8; ++j) {
        vals[j] = __expf(vals[j] - m);
        sum += vals[j];
    }
    red[t] = sum;
    __syncthreads();
    for (int s = 128; s > 0; s >>= 1) {
        if (t < s) red[t] += red[t + s];
        __syncthreads();
    }
    const float inv = 1.0f / red[0];
    __syncthreads();

    float* orow = out + (size_t)b * Ndim;
    #pragma unroll
    for (int j = 0; j < 8; ++j)
        orow[t + 256 * j] = vals[j] * inv;
}

// ---------------------------------------------------------------------------
extern "C" void kernel_launch(void* const* d_in, const int* in_sizes, int n_in,
                              void* d_out, int out_size, void* d_ws, size_t ws_size,
                              hipStream_t stream) {
    const float* feature = (const float*)d_in[0];   // [B, N, D]
    const float* hid     = (const float*)d_in[1];   // [B, D]
    const float* vvec    = (const float*)d_in[2];   // [1,1,D]
    const float* W       = (const float*)d_in[3];   // [1, D, 2D]
    float* out = (float*)d_out;                     // [B, 1, N]

    float* scores = (float*)d_ws;                                 // B*N floats
    float* cbd    = (float*)d_ws + (size_t)Bdim * Ndim;           // B*D floats

    precompute_c_kernel<<<Bdim, Ddim, 0, stream>>>(W, hid, cbd);
    attn_scores_kernel<<<1024, 256, 0, stream>>>(feature, W, vvec, cbd, scores);
    softmax_kernel<<<Bdim, 256, 0, stream>>>(scores, out);
}